// MultiheadAttention_v3_69303592288485
// MI455X (gfx1250) — compile-verified
//
#include <hip/hip_runtime.h>

typedef float v2f __attribute__((ext_vector_type(2)));
typedef float v8f __attribute__((ext_vector_type(8)));

#define SQ   512          // seq len
#define BB   16           // batch
#define EE   256          // embed
#define HH   8            // heads
#define HD   32           // head dim
#define HB   (HH*BB)      // 128 (h,b) pairs
#define ROWS (SQ*BB)      // 8192 token rows
#define KEEP 131072u      // (512*512)/2 kept per (h,b)

// ---- fp32 WMMA helper: D = A(16x4) * B(4x16) + C ------------------------
__device__ __forceinline__ v8f wmma_f32(v2f a, v2f b, v8f c) {
  return __builtin_amdgcn_wmma_f32_16x16x4_f32(
      /*neg_a=*/false, a, /*neg_b=*/false, b,
      /*c_mod=*/(short)0, c, /*reuse_a=*/false, /*reuse_b=*/false);
}

// monotonic float -> uint key (bigger key == bigger float)
__device__ __forceinline__ unsigned sortkey(float f) {
  unsigned u = __float_as_uint(f);
  return (u & 0x80000000u) ? ~u : (u | 0x80000000u);
}

// One 16x16 QK^T tile (unscaled). A-frag: lane = M (0..15), K = 2*(lane>>4)+j.
// B-frag: lane = N (k position), same K striping. Both arrays are (pos, d) major.
__device__ __forceinline__ v8f qk_tile(const float* __restrict__ qrow,
                                       const float* __restrict__ krow,
                                       int koff) {
  v8f acc = {0.f,0.f,0.f,0.f,0.f,0.f,0.f,0.f};
#pragma unroll
  for (int kk = 0; kk < HD; kk += 4) {
    v2f a = *(const v2f*)(qrow + kk + koff);
    v2f b = *(const v2f*)(krow + kk + koff);
    acc = wmma_f32(a, b, acc);
  }
  return acc;
}

// ======================= Kernel 1: LayerNorm + projection =================
// Y[(h*B+b)*S + s][d]  =  LN(x[row]) @ W + bias, row = s*B + b
__global__ __launch_bounds__(256) void lnproj_kernel(
    const float* __restrict__ x, const float* __restrict__ gamma,
    const float* __restrict__ beta, const float* __restrict__ W,
    const float* __restrict__ bias, float* __restrict__ Y) {
  __shared__ float at[16][258];         // 16 rows x 256 (padded, 8B-aligned rows)
  __shared__ float mu_s[16], rs_s[16];
  const int rt = blockIdx.x;            // 512 row tiles of 16
  for (int i = threadIdx.x; i < 16 * EE; i += 256) {
    int r = i >> 8, c = i & 255;
    at[r][c] = x[(size_t)(rt * 16 + r) * EE + c];
  }
  __syncthreads();
  if (threadIdx.x < 16) {
    int r = threadIdx.x;
    float s0 = 0.f;
    for (int c = 0; c < EE; ++c) s0 += at[r][c];
    float mu = s0 * (1.f / EE);
    float sq = 0.f;
    for (int c = 0; c < EE; ++c) { float d = at[r][c] - mu; sq += d * d; }
    mu_s[r] = mu;
    rs_s[r] = rsqrtf(sq * (1.f / EE) + 1e-5f);
  }
  __syncthreads();
  for (int i = threadIdx.x; i < 16 * EE; i += 256) {
    int r = i >> 8, c = i & 255;
    at[r][c] = (at[r][c] - mu_s[r]) * rs_s[r] * gamma[c] + beta[c];
  }
  __syncthreads();

  const int wv = threadIdx.x >> 5, lane = threadIdx.x & 31;
  const int koff = 2 * (lane >> 4);
  for (int nt = wv * 2; nt < wv * 2 + 2; ++nt) {   // each wave: 2 N-tiles
    v8f acc = {0.f,0.f,0.f,0.f,0.f,0.f,0.f,0.f};
    const int n = nt * 16 + (lane & 15);
#pragma unroll 4
    for (int kk = 0; kk < EE; kk += 4) {
      int k0 = kk + koff;
      v2f a = *(const v2f*)&at[lane & 15][k0];
      v2f b;
      b.x = W[(size_t)k0 * EE + n];
      b.y = W[(size_t)(k0 + 1) * EE + n];
      acc = wmma_f32(a, b, acc);
    }
    const float bv = bias[n];
    const int h = n >> 5, d = n & 31;
#pragma unroll
    for (int i = 0; i < 8; ++i) {
      int M = i + 8 * (lane >> 4);
      int g = rt * 16 + M;              // g = s*B + b  (B == 16)
      int s = g >> 4, bb = g & 15;
      Y[((size_t)(h * BB + bb) * SQ + s) * HD + d] = acc[i] + bv;
    }
  }
}

// ================= Kernel 2a: level-1 histogram + scan ====================
__global__ __launch_bounds__(256) void hist1_kernel(
    const float* __restrict__ Qh, const float* __restrict__ Kh,
    unsigned* __restrict__ bsel, unsigned* __restrict__ rnk) {
  __shared__ unsigned hist[4096];
  const int hb = blockIdx.x;
  for (int i = threadIdx.x; i < 4096; i += 256) hist[i] = 0u;
  __syncthreads();
  const int wv = threadIdx.x >> 5, lane = threadIdx.x & 31;
  const int koff = 2 * (lane >> 4);
  const float* Qb = Qh + (size_t)hb * SQ * HD;
  const float* Kb = Kh + (size_t)hb * SQ * HD;
  for (int t = wv; t < 1024; t += 8) {                 // 32x32 score tiles
    const float* qrow = Qb + (size_t)((t >> 5) * 16 + (lane & 15)) * HD;
    const float* krow = Kb + (size_t)((t & 31) * 16 + (lane & 15)) * HD;
    v8f acc = qk_tile(qrow, krow, koff);
#pragma unroll
    for (int i = 0; i < 8; ++i)
      atomicAdd(&hist[sortkey(acc[i] * 0.0625f) >> 20], 1u);
  }
  __syncthreads();
  if (threadIdx.x == 0) {
    unsigned need = KEEP, cum = 0;
    int b = 4095;
    for (; b > 0; --b) {
      unsigned h = hist[b];
      if (cum + h >= need) break;
      cum += h;
    }
    bsel[hb] = (unsigned)b;
    rnk[hb] = need - cum;                // rank within the selected bin, >= 1
  }
}

// ================= Kernel 2b: level-2 refinement ==========================
__global__ __launch_bounds__(256) void hist2_kernel(
    const float* __restrict__ Qh, const float* __restrict__ Kh,
    const unsigned* __restrict__ bsel, const unsigned* __restrict__ rnk,
    unsigned* __restrict__ tkey) {
  __shared__ unsigned hist[4096];
  const int hb = blockIdx.x;
  for (int i = threadIdx.x; i < 4096; i += 256) hist[i] = 0u;
  __syncthreads();
  const unsigned b1 = bsel[hb];
  const int wv = threadIdx.x >> 5, lane = threadIdx.x & 31;
  const int koff = 2 * (lane >> 4);
  const float* Qb = Qh + (size_t)hb * SQ * HD;
  const float* Kb = Kh + (size_t)hb * SQ * HD;
  for (int t = wv; t < 1024; t += 8) {
    const float* qrow = Qb + (size_t)((t >> 5) * 16 + (lane & 15)) * HD;
    const float* krow = Kb + (size_t)((t & 31) * 16 + (lane & 15)) * HD;
    v8f acc = qk_tile(qrow, krow, koff);
#pragma unroll
    for (int i = 0; i < 8; ++i) {
      unsigned k = sortkey(acc[i] * 0.0625f);
      if ((k >> 20) == b1) atomicAdd(&hist[(k >> 8) & 0xFFFu], 1u);
    }
  }
  __syncthreads();
  if (threadIdx.x == 0) {
    unsigned need = rnk[hb], cum = 0;
    int b = 4095;
    for (; b > 0; --b) {
      unsigned h = hist[b];
      if (cum + h >= need) break;
      cum += h;
    }
    tkey[hb] = (b1 << 20) | ((unsigned)b << 8);   // keep if key >= tkey
  }
}

// ============ Kernel 3: masked softmax-attention (flash-style) ============
// One wave per (hb, qtile). exp(-1e9)==0 and kept scores are O(1), so a
// plain (no running max) exp/sum is numerically safe.
__global__ __launch_bounds__(256) void attn_kernel(
    const float* __restrict__ Qh, const float* __restrict__ Kh,
    const float* __restrict__ Vh, const unsigned* __restrict__ tkey,
    float* __restrict__ AO /* (S,B,E) */) {
  __shared__ float pbuf[8][16][18];     // per-wave P staging (D -> A relayout)
  const int wv = threadIdx.x >> 5, lane = threadIdx.x & 31;
  const int w = blockIdx.x * 8 + wv;    // 4096 wave tasks
  const int hb = w >> 5, qtile = w & 31;
  const unsigned T = tkey[hb];
  const int koff = 2 * (lane >> 4);
  const float* Qb = Qh + (size_t)hb * SQ * HD;
  const float* Kb = Kh + (size_t)hb * SQ * HD;
  const float* Vb = Vh + (size_t)hb * SQ * HD;
  const float* qrow = Qb + (size_t)(qtile * 16 + (lane & 15)) * HD;

  v8f o0 = {0.f,0.f,0.f,0.f,0.f,0.f,0.f,0.f};   // d = 0..15
  v8f o1 = {0.f,0.f,0.f,0.f,0.f,0.f,0.f,0.f};   // d = 16..31
  float rs[8] = {0.f,0.f,0.f,0.f,0.f,0.f,0.f,0.f};

  for (int kt = 0; kt < 32; ++kt) {
    const float* krow = Kb + (size_t)(kt * 16 + (lane & 15)) * HD;
    v8f acc = qk_tile(qrow, krow, koff);
    float p[8];
#pragma unroll
    for (int i = 0; i < 8; ++i) {
      float s = acc[i] * 0.0625f;
      float pv = (sortkey(s) >= T) ? __expf(s) : 0.f;
      p[i] = pv;
      rs[i] += pv;
    }
    // D-layout (M = i + 8*(lane>>4), N = lane&15) -> LDS
    __builtin_amdgcn_wave_barrier();
#pragma unroll
    for (int i = 0; i < 8; ++i)
      pbuf[wv][i + 8 * (lane >> 4)][lane & 15] = p[i];
    __builtin_amdgcn_wave_barrier();
    // O += P(16x16) @ V(16x32): 4 K-steps x 2 d-halves
#pragma unroll
    for (int dh = 0; dh < 2; ++dh) {
      const int d = dh * 16 + (lane & 15);
      v8f oacc = dh ? o1 : o0;
#pragma unroll
      for (int ks = 0; ks < 4; ++ks) {
        const int k0 = ks * 4 + koff;
        v2f a = *(const v2f*)&pbuf[wv][lane & 15][k0];
        v2f b;
        b.x = Vb[(size_t)(kt * 16 + k0) * HD + d];
        b.y = Vb[(size_t)(kt * 16 + k0 + 1) * HD + d];
        oacc = wmma_f32(a, b, oacc);
      }
      if (dh) o1 = oacc; else o0 = oacc;
    }
    __builtin_amdgcn_wave_barrier();
  }
  // reduce row sums across the 16 lanes of each half (rows i+8*(lane>>4))
#pragma unroll
  for (int i = 0; i < 8; ++i) {
    float v = rs[i];
    for (int m = 1; m < 16; m <<= 1) v += __shfl_xor(v, m, 32);
    rs[i] = v;
  }
  const int h = hb >> 4, b = hb & 15;
#pragma unroll
  for (int i = 0; i < 8; ++i) {
    int M = i + 8 * (lane >> 4);
    int s = qtile * 16 + M;
    float inv = rs[i] > 0.f ? 1.f / rs[i] : 0.f;
    int d0 = lane & 15;
    size_t base = ((size_t)s * BB + b) * EE + h * HD;
    AO[base + d0]      = o0[i] * inv;
    AO[base + 16 + d0] = o1[i] * inv;
  }
}

// =================== Kernel 4: output projection ==========================
__global__ __launch_bounds__(256) void proj_kernel(
    const float* __restrict__ X, const float* __restrict__ W,
    const float* __restrict__ bias, float* __restrict__ Y) {
  __shared__ float at[16][258];
  const int rt = blockIdx.x;
  for (int i = threadIdx.x; i < 16 * EE; i += 256) {
    int r = i >> 8, c = i & 255;
    at[r][c] = X[(size_t)(rt * 16 + r) * EE + c];
  }
  __syncthreads();
  const int wv = threadIdx.x >> 5, lane = threadIdx.x & 31;
  const int koff = 2 * (lane >> 4);
  for (int nt = wv * 2; nt < wv * 2 + 2; ++nt) {
    v8f acc = {0.f,0.f,0.f,0.f,0.f,0.f,0.f,0.f};
    const int n = nt * 16 + (lane & 15);
#pragma unroll 4
    for (int kk = 0; kk < EE; kk += 4) {
      int k0 = kk + koff;
      v2f a = *(const v2f*)&at[lane & 15][k0];
      v2f b;
      b.x = W[(size_t)k0 * EE + n];
      b.y = W[(size_t)(k0 + 1) * EE + n];
      acc = wmma_f32(a, b, acc);
    }
    const float bv = bias[n];
#pragma unroll
    for (int i = 0; i < 8; ++i) {
      int M = i + 8 * (lane >> 4);
      Y[(size_t)(rt * 16 + M) * EE + n] = acc[i] + bv;
    }
  }
}

// ============================== launcher ==================================
extern "C" void kernel_launch(void* const* d_in, const int* in_sizes, int n_in,
                              void* d_out, int out_size, void* d_ws,
                              size_t ws_size, hipStream_t stream) {
  (void)in_sizes; (void)n_in; (void)out_size; (void)ws_size;
  const float* q    = (const float*)d_in[0];
  const float* k    = (const float*)d_in[1];
  const float* v    = (const float*)d_in[2];
  const float* ln_g = (const float*)d_in[3];
  const float* ln_b = (const float*)d_in[4];
  const float* wq   = (const float*)d_in[5];
  const float* bq   = (const float*)d_in[6];
  const float* wk   = (const float*)d_in[7];
  const float* bk   = (const float*)d_in[8];
  const float* wv   = (const float*)d_in[9];
  const float* bv   = (const float*)d_in[10];
  const float* wo   = (const float*)d_in[11];
  const float* bo   = (const float*)d_in[12];

  const size_t TN = (size_t)ROWS * EE;      // 2,097,152 floats per tensor
  float* Qh = (float*)d_ws;
  float* Kh = Qh + TN;
  float* Vh = Kh + TN;
  float* AO = Vh + TN;
  unsigned* bsel = (unsigned*)(AO + TN);
  unsigned* rnk  = bsel + HB;
  unsigned* tkey = rnk + HB;

  lnproj_kernel<<<ROWS / 16, 256, 0, stream>>>(q, ln_g, ln_b, wq, bq, Qh);
  lnproj_kernel<<<ROWS / 16, 256, 0, stream>>>(k, ln_g, ln_b, wk, bk, Kh);
  lnproj_kernel<<<ROWS / 16, 256, 0, stream>>>(v, ln_g, ln_b, wv, bv, Vh);
  hist1_kernel<<<HB, 256, 0, stream>>>(Qh, Kh, bsel, rnk);
  hist2_kernel<<<HB, 256, 0, stream>>>(Qh, Kh, bsel, rnk, tkey);
  attn_kernel<<<(HB * 32) / 8, 256, 0, stream>>>(Qh, Kh, Vh, tkey, AO);
  proj_kernel<<<ROWS / 16, 256, 0, stream>>>(AO, wo, bo, (float*)d_out);
}